// GSRNet_4191888081056
// MI455X (gfx1250) — compile-verified
//
#include <hip/hip_runtime.h>
#include <math.h>

typedef float v2f __attribute__((ext_vector_type(2)));
typedef float v8f __attribute__((ext_vector_type(8)));

#define GF_TA  1
#define GF_TB  2
#define GF_ABS 4

// ---------------------------------------------------------------------------
// Generic f32 WMMA GEMM: C = op(A) @ op(B) [+bias(row)] [+addend] [abs]
// One wave (32 lanes) computes one 16x16 f32 tile via V_WMMA_F32_16X16X4_F32.
// A frag (16x4): lane l, vgpr v -> M = l&15, K = v + 2*(l>>4)
// B frag (4x16): lane l, vgpr v -> N = l&15, K = v + 2*(l>>4)
// C/D (16x16):   lane l, vgpr r -> M = r + 8*(l>>4), N = l&15
//
// TA/TB are template params so the k-loop has zero branches. M/N edges need
// no masking in the mainloop: garbage A-rows / B-cols only affect D entries
// that the epilogue never stores. Only K-tail (K%4!=0) is masked.
// Contiguous k-pairs are loaded as aligned v2f (global_load_b64): all buffers
// are 8B-aligned, all leading dims are even, and kk = k0+2*half is even.
// ---------------------------------------------------------------------------
template <bool TA, bool TB>
__global__ __launch_bounds__(32)
void gemm_wmma_f32_t(const float* __restrict__ A, int lda,
                     const float* __restrict__ B, int ldb,
                     const float* __restrict__ bias,
                     const float* __restrict__ addend,
                     float* __restrict__ C, int ldc,
                     int M, int N, int K, int doabs)
{
    const int lane = threadIdx.x & 31;
    const int half = lane >> 4;
    const int l16  = lane & 15;
    const int m0 = blockIdx.y * 16;
    const int n0 = blockIdx.x * 16;

    // clamp OOB lanes into range (their products land in OOB D rows/cols only)
    const int am = (m0 + l16 < M) ? (m0 + l16) : 0;
    const int bn = (n0 + l16 < N) ? (n0 + l16) : 0;

    const float* ApBase = TA ? (A + am) : (A + (size_t)am * lda);
    const float* BpBase = TB ? (B + (size_t)bn * ldb) : (B + bn);
    const int k2 = 2 * half;

    v8f acc = {0.f,0.f,0.f,0.f,0.f,0.f,0.f,0.f};

    int k0 = 0;
#pragma unroll 4
    for (; k0 + 4 <= K; k0 += 4) {
        const int kk = k0 + k2;
        v2f af, bf;
        if (TA) { af[0] = ApBase[(size_t)kk * lda];
                  af[1] = ApBase[(size_t)(kk + 1) * lda]; }
        else    { af = *(const v2f*)(ApBase + kk); }        // b64 load
        if (TB) { bf = *(const v2f*)(BpBase + kk); }        // b64 load
        else    { bf[0] = BpBase[(size_t)kk * ldb];
                  bf[1] = BpBase[(size_t)(kk + 1) * ldb]; }
        acc = __builtin_amdgcn_wmma_f32_16x16x4_f32(
                false, af, false, bf, (short)0, acc, false, false);
    }
    if (k0 < K) {   // K % 4 tail (only K=134 hits this)
        const int kk = k0 + k2;
        v2f af = {0.f, 0.f}, bf = {0.f, 0.f};
#pragma unroll
        for (int v = 0; v < 2; ++v) {
            if (kk + v < K) {
                af[v] = TA ? ApBase[(size_t)(kk + v) * lda] : ApBase[kk + v];
                bf[v] = TB ? BpBase[kk + v] : BpBase[(size_t)(kk + v) * ldb];
            }
        }
        acc = __builtin_amdgcn_wmma_f32_16x16x4_f32(
                false, af, false, bf, (short)0, acc, false, false);
    }

#pragma unroll
    for (int r = 0; r < 8; ++r) {
        const int m = m0 + r + 8 * half;
        const int n = n0 + l16;
        if (m < M && n < N) {
            float v = acc[r];
            if (bias)   v += bias[n];
            if (addend) v += addend[(size_t)m * ldc + n];
            if (doabs)  v = fabsf(v);
            C[(size_t)m * ldc + n] = v;
        }
    }
}

// ---------------------------------------------------------------------------
// Small helper kernels
// ---------------------------------------------------------------------------
__global__ void k_startx(const float* __restrict__ w, const float* __restrict__ b,
                         float* __restrict__ org, float* __restrict__ out_start,
                         int rows, int cols)
{
    int t = blockIdx.x * blockDim.x + threadIdx.x;
    if (t < rows * cols) {
        float v = w[t] + b[t % cols];
        org[t] = v;
        out_start[t] = v;
    }
}

// r[i] = rowsum(M[i,:])^(-1/2), inf -> 0
__global__ void k_rowsum_rsqrt(const float* __restrict__ M, float* __restrict__ r,
                               int rows, int cols)
{
    int i = blockIdx.x * blockDim.x + threadIdx.x;
    if (i < rows) {
        float s = 0.f;
        for (int j = 0; j < cols; ++j) s += M[(size_t)i * cols + j];
        r[i] = (s > 0.f) ? rsqrtf(s) : 0.f;
    }
}

// out[i,j] = M[j,i] * r[i] * r[j]   ( == (M * r).T * r )
__global__ void k_norm_apply(const float* __restrict__ M, const float* __restrict__ r,
                             float* __restrict__ out, int n)
{
    int t = blockIdx.x * blockDim.x + threadIdx.x;
    if (t < n * n) {
        int i = t / n, j = t % n;
        out[t] = M[(size_t)j * n + i] * r[i] * r[j];
    }
}

__global__ void k_pool_scores(const float* __restrict__ X, const float* __restrict__ pw,
                              const float* __restrict__ pb, float* __restrict__ scores,
                              int n, int dim)
{
    int i = blockIdx.x * blockDim.x + threadIdx.x;
    if (i < n) {
        float s = 0.f;
        for (int c = 0; c < dim; ++c) s += X[(size_t)i * dim + c] * pw[c];
        s = (s + pb[0]) * 0.01f;
        scores[i] = 1.f / (1.f + expf(-s));
    }
}

// deterministic top-k, descending, lowest-index tie-break (matches lax.top_k)
__global__ void k_topk(const float* __restrict__ scores, int n, int k,
                       int* __restrict__ idx, float* __restrict__ vals)
{
    if (blockIdx.x == 0 && threadIdx.x == 0) {
        bool used[300];
        for (int i = 0; i < n; ++i) used[i] = false;
        for (int r = 0; r < k; ++r) {
            int bi = -1; float bv = -1e30f;
            for (int i = 0; i < n; ++i)
                if (!used[i] && scores[i] > bv) { bv = scores[i]; bi = i; }
            used[bi] = true;
            idx[r] = bi;
            vals[r] = bv;
        }
    }
}

__global__ void k_gather_scale(const float* __restrict__ X, const int* __restrict__ idx,
                               const float* __restrict__ vals, float* __restrict__ out,
                               int k, int cols)
{
    int t = blockIdx.x * blockDim.x + threadIdx.x;
    if (t < k * cols) {
        int r = t / cols, c = t % cols;
        out[t] = X[(size_t)idx[r] * cols + c] * vals[r];
    }
}

__global__ void k_fill(float* __restrict__ p, int n, float v)
{
    int t = blockIdx.x * blockDim.x + threadIdx.x;
    if (t < n) p[t] = v;
}

__global__ void k_scatter_rows(const float* __restrict__ X, const int* __restrict__ idx,
                               float* __restrict__ out, int k, int cols)
{
    int t = blockIdx.x * blockDim.x + threadIdx.x;
    if (t < k * cols) {
        int r = t / cols, c = t % cols;
        out[(size_t)idx[r] * cols + c] = X[t];
    }
}

__global__ void k_concat(const float* __restrict__ X, const float* __restrict__ org,
                         float* __restrict__ out, int rows, int cols)
{
    int t = blockIdx.x * blockDim.x + threadIdx.x;
    int w = 2 * cols;
    if (t < rows * w) {
        int i = t / w, c = t % w;
        out[t] = (c < cols) ? X[(size_t)i * cols + c] : org[(size_t)i * cols + (c - cols)];
    }
}

// a[i,j] = gsr_w[i, j] + gsr_w[i, j + LR]
__global__ void k_build_a(const float* __restrict__ gw, float* __restrict__ a,
                          int hr, int lr)
{
    int t = blockIdx.x * blockDim.x + threadIdx.x;
    if (t < hr * lr) {
        int i = t / lr, j = t % lr;
        a[t] = gw[(size_t)i * hr + j] + gw[(size_t)i * hr + lr + j];
    }
}

// Asym = triu(A) + triu(A,1).T
__global__ void k_sym_upper(const float* __restrict__ A, float* __restrict__ out, int n)
{
    int t = blockIdx.x * blockDim.x + threadIdx.x;
    if (t < n * n) {
        int i = t / n, j = t % n;
        out[t] = (j >= i) ? A[(size_t)i * n + j] : A[(size_t)j * n + i];
    }
}

__global__ void k_diag1(float* __restrict__ M, int n)
{
    int i = blockIdx.x * blockDim.x + threadIdx.x;
    if (i < n) M[(size_t)i * n + i] = 1.0f;
}

// out[i,j] = (i==j) ? 1 : |0.5*(S[i,j]+S[j,i])|
__global__ void k_symdiag1abs(const float* __restrict__ S, float* __restrict__ out, int n)
{
    int t = blockIdx.x * blockDim.x + threadIdx.x;
    if (t < n * n) {
        int i = t / n, j = t % n;
        out[t] = (i == j) ? 1.0f
                          : fabsf(0.5f * (S[(size_t)i * n + j] + S[(size_t)j * n + i]));
    }
}

// ---------------------------------------------------------------------------
// Single-block cyclic Jacobi eigensolver (symmetric, ascending eigenvalues).
// Aw destroyed; V workspace; U = eigenvectors sorted ascending by eigenvalue.
// ---------------------------------------------------------------------------
__global__ __launch_bounds__(256)
void k_jacobi_eigh(float* __restrict__ Aw, float* __restrict__ V,
                   float* __restrict__ U, int n, int sweeps)
{
    const int tid = threadIdx.x;
    __shared__ float sc, ss;
    __shared__ int sskip;
    __shared__ short perm[300];

    for (int t = tid; t < n * n; t += blockDim.x)
        V[t] = (t / n == t % n) ? 1.0f : 0.0f;
    __syncthreads();

    for (int sw = 0; sw < sweeps; ++sw) {
        for (int p = 0; p < n - 1; ++p) {
            for (int q = p + 1; q < n; ++q) {
                if (tid == 0) {
                    float apq = Aw[p * n + q];
                    float app = Aw[p * n + p];
                    float aqq = Aw[q * n + q];
                    if (fabsf(apq) <= 1e-20f) {
                        sskip = 1;
                    } else {
                        sskip = 0;
                        float theta = (aqq - app) / (2.0f * apq);
                        float tt = ((theta >= 0.f) ? 1.f : -1.f) /
                                   (fabsf(theta) + sqrtf(theta * theta + 1.f));
                        float c = 1.f / sqrtf(tt * tt + 1.f);
                        float s = tt * c;
                        sc = c; ss = s;
                        Aw[p * n + p] = app - tt * apq;
                        Aw[q * n + q] = aqq + tt * apq;
                        Aw[p * n + q] = 0.f;
                        Aw[q * n + p] = 0.f;
                    }
                }
                __syncthreads();
                if (!sskip) {
                    float c = sc, s = ss;
                    for (int j = tid; j < n; j += blockDim.x) {
                        if (j != p && j != q) {
                            float ajp = Aw[j * n + p], ajq = Aw[j * n + q];
                            float np_ = c * ajp - s * ajq;
                            float nq_ = s * ajp + c * ajq;
                            Aw[j * n + p] = np_; Aw[p * n + j] = np_;
                            Aw[j * n + q] = nq_; Aw[q * n + j] = nq_;
                        }
                    }
                    for (int j = tid; j < n; j += blockDim.x) {
                        float vjp = V[j * n + p], vjq = V[j * n + q];
                        V[j * n + p] = c * vjp - s * vjq;
                        V[j * n + q] = s * vjp + c * vjq;
                    }
                }
                __syncthreads();
            }
        }
    }

    if (tid == 0) {   // selection sort of eigenvalues, ascending
        bool used[300];
        for (int i = 0; i < n; ++i) used[i] = false;
        for (int r = 0; r < n; ++r) {
            int bi = -1; float bv = 0.f;
            for (int i = 0; i < n; ++i) {
                if (!used[i]) {
                    float ev = Aw[i * n + i];
                    if (bi < 0 || ev < bv) { bv = ev; bi = i; }
                }
            }
            used[bi] = true;
            perm[r] = (short)bi;
        }
    }
    __syncthreads();
    for (int t = tid; t < n * n; t += blockDim.x) {
        int i = t / n, j = t % n;
        U[(size_t)i * n + j] = V[(size_t)i * n + perm[j]];
    }
}

// ---------------------------------------------------------------------------
// Host orchestration
// ---------------------------------------------------------------------------
static inline int cdiv(int a, int b) { return (a + b - 1) / b; }

extern "C" void kernel_launch(void* const* d_in, const int* in_sizes, int n_in,
                              void* d_out, int out_size, void* d_ws, size_t ws_size,
                              hipStream_t stream)
{
    (void)in_sizes; (void)n_in; (void)out_size; (void)ws_size;

    const int LR = 134, HR = 268, DIM = 320;
    const int ns[5] = {134, 120, 84, 50, 25};

    const float* lr       = (const float*)d_in[0];
    const float* start_w  = (const float*)d_in[1];
    const float* start_b  = (const float*)d_in[2];
    const float* down_w   = (const float*)d_in[3];   // (4,320,320)
    const float* down_b   = (const float*)d_in[4];   // (4,320)
    const float* pool_w   = (const float*)d_in[5];   // (4,320,1)
    const float* pool_b   = (const float*)d_in[6];   // (4,1)
    const float* bottom_w = (const float*)d_in[7];
    const float* bottom_b = (const float*)d_in[8];
    const float* up_w     = (const float*)d_in[9];
    const float* up_b     = (const float*)d_in[10];
    const float* end_w    = (const float*)d_in[11];  // (640,268)
    const float* end_b    = (const float*)d_in[12];
    const float* gsr_w    = (const float*)d_in[13];  // (268,268)
    const float* gc1_w    = (const float*)d_in[14];
    const float* gc2_w    = (const float*)d_in[15];

    float* out       = (float*)d_out;
    float* z_out     = out;                                 // 268*268
    float* net_outs  = out + HR * HR;                       // 134*268
    float* start_out = out + HR * HR + LR * HR;             // 134*320
    float* adj_out   = out + HR * HR + LR * HR + LR * DIM;  // 268*268

    // bump allocator over workspace (floats)
    float* wsp = (float*)d_ws;
    auto alloc = [&](size_t nfl) { float* p = wsp; wsp += nfl; return p; };

    float* org    = alloc((size_t)LR * DIM);
    float* A0     = alloc((size_t)LR * LR);
    float* r1     = alloc(LR);
    float* downs[4];
    for (int i = 0; i < 4; ++i) downs[i] = alloc((size_t)LR * DIM);
    float* scores = alloc(LR);
    float* valsb  = alloc(4 * 134);
    int*   idxb   = (int*)alloc(4 * 134);
    float* Xa     = alloc((size_t)LR * 2 * DIM);  // also used for concat (134x640)
    float* Xb     = alloc((size_t)LR * 2 * DIM);
    float* Amat   = alloc((size_t)LR * LR);
    float* Vm     = alloc((size_t)LR * LR);
    float* Um     = alloc((size_t)LR * LR);
    float* Tm     = alloc((size_t)LR * HR);
    float* amat   = alloc((size_t)HR * LR);
    float* fd     = alloc((size_t)HR * HR);
    float* r2     = alloc(HR);
    float* Zraw   = alloc((size_t)HR * HR);
    float* Zm     = alloc((size_t)HR * HR);
    float* Hm     = alloc((size_t)HR * HR);
    float* Gm     = alloc((size_t)HR * HR);

    auto gemm = [&](const float* A, int lda, const float* B, int ldb,
                    const float* bias, const float* addend,
                    float* C, int ldc, int M, int N, int K, int flags) {
        dim3 g(cdiv(N, 16), cdiv(M, 16));
        int doabs = (flags & GF_ABS) ? 1 : 0;
        if (flags & GF_TA)
            gemm_wmma_f32_t<true, false><<<g, 32, 0, stream>>>(
                A, lda, B, ldb, bias, addend, C, ldc, M, N, K, doabs);
        else if (flags & GF_TB)
            gemm_wmma_f32_t<false, true><<<g, 32, 0, stream>>>(
                A, lda, B, ldb, bias, addend, C, ldc, M, N, K, doabs);
        else
            gemm_wmma_f32_t<false, false><<<g, 32, 0, stream>>>(
                A, lda, B, ldb, bias, addend, C, ldc, M, N, K, doabs);
    };

    // ---- start: X0 = start_w + start_b ; start_outs = X0 ----
    k_startx<<<cdiv(LR * DIM, 256), 256, 0, stream>>>(start_w, start_b, org,
                                                      start_out, LR, DIM);

    // ---- A0 = normalize_adj(lr) ----
    k_rowsum_rsqrt<<<cdiv(LR, 128), 128, 0, stream>>>(lr, r1, LR, LR);
    k_norm_apply<<<cdiv(LR * LR, 256), 256, 0, stream>>>(lr, r1, A0, LR);

    // ---- down path ----
    const float* Xprev = org;
    for (int i = 0; i < 4; ++i) {
        int n = ns[i], k = ns[i + 1];
        gemm(Xprev, DIM, down_w + (size_t)i * DIM * DIM, DIM,
             down_b + (size_t)i * DIM, nullptr, downs[i], DIM, n, DIM, DIM, 0);
        k_pool_scores<<<cdiv(n, 128), 128, 0, stream>>>(
            downs[i], pool_w + (size_t)i * DIM, pool_b + i, scores, n, DIM);
        k_topk<<<1, 1, 0, stream>>>(scores, n, k, idxb + i * 134, valsb + i * 134);
        k_gather_scale<<<cdiv(k * DIM, 256), 256, 0, stream>>>(
            downs[i], idxb + i * 134, valsb + i * 134, Xa, k, DIM);
        Xprev = Xa;
    }

    // ---- bottom ----
    gemm(Xa, DIM, bottom_w, DIM, bottom_b, nullptr, Xb, DIM, ns[4], DIM, DIM, 0);

    // ---- up path (adjacency values unused: only shapes + indices matter) ----
    for (int i = 0; i < 4; ++i) {
        int j = 3 - i;
        int rows_in = ns[j + 1], rows_out = ns[j];
        k_fill<<<cdiv(rows_out * DIM, 256), 256, 0, stream>>>(Xa, rows_out * DIM, 0.f);
        k_scatter_rows<<<cdiv(rows_in * DIM, 256), 256, 0, stream>>>(
            Xb, idxb + j * 134, Xa, rows_in, DIM);
        gemm(Xa, DIM, up_w + (size_t)i * DIM * DIM, DIM,
             up_b + (size_t)i * DIM, downs[j], Xb, DIM, rows_out, DIM, DIM, 0);
    }

    // ---- end: net_outs = [X, org] @ end_w + end_b ----
    k_concat<<<cdiv(LR * 2 * DIM, 256), 256, 0, stream>>>(Xb, org, Xa, LR, DIM);
    gemm(Xa, 2 * DIM, end_w, HR, end_b, nullptr, net_outs, HR, LR, HR, 2 * DIM, 0);

    // ---- GSR: eigendecomposition of symmetrized A0 ----
    k_sym_upper<<<cdiv(LR * LR, 256), 256, 0, stream>>>(A0, Amat, LR);
    k_jacobi_eigh<<<1, 256, 0, stream>>>(Amat, Vm, Um, LR, 12);

    // T = U^T @ net_outs  (134x268)
    gemm(Um, LR, net_outs, HR, nullptr, nullptr, Tm, HR, LR, HR, LR, GF_TA);

    // a = gsr_w[:, :134] + gsr_w[:, 134:]  (268x134); f_d = |a @ T|, diag<-1
    k_build_a<<<cdiv(HR * LR, 256), 256, 0, stream>>>(gsr_w, amat, HR, LR);
    gemm(amat, LR, Tm, HR, nullptr, nullptr, fd, HR, HR, HR, LR, GF_ABS);
    k_diag1<<<cdiv(HR, 128), 128, 0, stream>>>(fd, HR);

    // adj = normalize_adj(f_d)  -> "outputs"
    k_rowsum_rsqrt<<<cdiv(HR, 128), 128, 0, stream>>>(fd, r2, HR, HR);
    k_norm_apply<<<cdiv(HR * HR, 256), 256, 0, stream>>>(fd, r2, adj_out, HR);

    // Z = sym(adj @ adj^T), diag<-1, abs
    gemm(adj_out, HR, adj_out, HR, nullptr, nullptr, Zraw, HR, HR, HR, HR, GF_TB);
    k_symdiag1abs<<<cdiv(HR * HR, 256), 256, 0, stream>>>(Zraw, Zm, HR);

    // h1 = adj @ (Z @ gc1_w); h2 = adj @ (h1 @ gc2_w)
    gemm(Zm, HR, gc1_w, HR, nullptr, nullptr, Hm, HR, HR, HR, HR, 0);
    gemm(adj_out, HR, Hm, HR, nullptr, nullptr, Gm, HR, HR, HR, HR, 0);
    gemm(Gm, HR, gc2_w, HR, nullptr, nullptr, Hm, HR, HR, HR, HR, 0);
    gemm(adj_out, HR, Hm, HR, nullptr, nullptr, Zraw, HR, HR, HR, HR, 0);

    // z = |sym(h2) with diag 1|
    k_symdiag1abs<<<cdiv(HR * HR, 256), 256, 0, stream>>>(Zraw, z_out, HR);
}